// Spatial_Mask_48936857370961
// MI455X (gfx1250) — compile-verified
//
#include <hip/hip_runtime.h>

typedef __attribute__((ext_vector_type(2))) float v2f;
typedef __attribute__((ext_vector_type(8))) float v8f;

// Problem constants (from reference): B=64, C=64, H=W=128, mask 64x64 -> out 128x128
#define BSZ   64
#define CCH   64
#define HH    128
#define WW    128
#define PLANE (HH * WW)        // 16384 elems per (b,c) plane

// Each wave handles one tile: batch b, pooled row hp, 16 pooled cols (wt*16 .. wt*16+15).
// z[m] = sum_c dw[c] * (2x2 block sum of x at pooled pixel (hp, wt*16+m))
// mask = (0.25*z + db >= 0), written as 2x2 nearest-upsampled pixels.
__global__ __launch_bounds__(256) void spatial_mask_wmma_kernel(
    const float* __restrict__ x,     // [64][64][128][128]
    const float* __restrict__ w,     // [2][64]
    const float* __restrict__ bias,  // [2]
    float* __restrict__ out)         // [64][128][128]
{
    __shared__ float dw[CCH];        // w[0,c] - w[1,c]
    __shared__ float zsh[8][16];     // per-wave transpose buffer

    const int tid = threadIdx.x;
    if (tid < CCH) dw[tid] = w[tid] - w[CCH + tid];
    __syncthreads();

    const float db = bias[0] - bias[1];

    const int wv   = tid >> 5;       // wave id in block (0..7)
    const int lane = tid & 31;

    const int tile = blockIdx.x * 8 + wv;       // 0 .. 16383
    const int wt   = tile & 3;                  // 4 col-tiles of 16 pooled cols
    const int hp   = (tile >> 2) & 63;          // pooled row 0..63
    const int b    = tile >> 8;                 // batch 0..63

    const int m    = lane & 15;                 // spatial position within tile (M row of A)
    const int koff = (lane >> 4) * 2;           // lanes 0-15: K=0,1 ; lanes 16-31: K=2,3

    // x[b][0][2*hp][wt*32 + 2*m] : each A-element needs a 2x2 f32 block here
    const float* xb = x + (size_t)b * (CCH * PLANE)
                        + (size_t)(2 * hp) * WW + wt * 32 + 2 * m;

    v8f acc = {};
    #pragma unroll
    for (int s = 0; s < 16; ++s) {              // K-steps: 16 * K4 = C=64 channels
        const int c0 = 4 * s + koff;
        v2f a, bv;
        #pragma unroll
        for (int j = 0; j < 2; ++j) {
            const float* p = xb + (size_t)(c0 + j) * PLANE;
            v2f r0 = *(const v2f*)p;            // row 2*hp,   2 pixels (8B, coalesced 128B/half-wave)
            v2f r1 = *(const v2f*)(p + WW);     // row 2*hp+1, 2 pixels
            a[j]  = (r0.x + r0.y) + (r1.x + r1.y);  // 2x2 block sum (scale 0.25 applied later)
            bv[j] = dw[c0 + j];                 // broadcast dw across all 16 N columns
        }
        // D = A(16x4 f32) * B(4x16 f32) + C ; every N column identical by construction
        acc = __builtin_amdgcn_wmma_f32_16x16x4_f32(
            /*neg_a=*/false, a, /*neg_b=*/false, bv,
            /*c_mod=*/(short)0, acc, /*reuse_a=*/false, /*reuse_b=*/false);
    }

    // C/D layout: VGPR j -> M=j (lanes 0-15) / M=j+8 (lanes 16-31); columns are duplicates.
    // Lane 0 holds z[0..7], lane 16 holds z[8..15] -> stage through LDS to fan out.
    if ((lane & 15) == 0) {
        const int base = (lane >> 4) * 8;
        #pragma unroll
        for (int j = 0; j < 8; ++j) zsh[wv][base + j] = acc[j];
    }
    __syncthreads();

    const float z  = zsh[wv][lane & 15];
    const float mv = (0.25f * z + db >= 0.0f) ? 1.0f : 0.0f;

    // Nearest 2x upsample: lanes 0-15 write row 2*hp, lanes 16-31 write row 2*hp+1;
    // each lane writes a float2 (two duplicated cols). 128B contiguous per half-wave.
    float* o = out + (size_t)b * PLANE
                   + (size_t)(2 * hp + (lane >> 4)) * WW
                   + wt * 32 + 2 * (lane & 15);
    v2f ov; ov.x = mv; ov.y = mv;
    *(v2f*)o = ov;
}

extern "C" void kernel_launch(void* const* d_in, const int* in_sizes, int n_in,
                              void* d_out, int out_size, void* d_ws, size_t ws_size,
                              hipStream_t stream) {
    const float* x    = (const float*)d_in[0];  // [64,64,128,128] f32
    const float* w    = (const float*)d_in[1];  // [2,64] f32
    const float* bias = (const float*)d_in[2];  // [2] f32
    // d_in[3] = temperature (unused in eval branch)
    float* out = (float*)d_out;                 // [64,1,128,128] f32

    // 16384 wave-tiles / 8 waves per block
    hipLaunchKernelGGL(spatial_mask_wmma_kernel, dim3(2048), dim3(256), 0, stream,
                       x, w, bias, out);
}